// RandomAdjustTemp_9380208574640
// MI455X (gfx1250) — compile-verified
//
#include <hip/hip_runtime.h>

// Per-channel white-balance multiply, kelvin 5500K:
//   scale = {255/255, 236/255, 224/255} for channels {R,G,B}
// Tensor layout: (B=64, C=3, H=512, W=512) fp32 contiguous.
// Channel stride = 512*512 = 262144 floats = 65536 float4s.
//
// Memory-bound (402 MB traffic -> ~17 us HBM floor @ 23.3 TB/s).
// Strategy:
//  - straight-line full-tile kernel: no bounds checks, no exec-mask branches
//  - 4 back-to-back b128 loads with TH=NT (streaming, zero reuse, don't
//    pollute the 192 MB L2), then 16 v_mul_f32, then 4 b128 NT stores
//  - scale factor is wave-uniform (block tile of 1024 float4s divides the
//    65536-float4 channel period) -> computed on the SALU from blockIdx
//  - scalar tail kernel for generality (not launched at this size)

typedef float v4f __attribute__((ext_vector_type(4)));

#define ITEMS_PER_THREAD 4
#define BLOCK_THREADS 256
#define TILE_F4 (BLOCK_THREADS * ITEMS_PER_THREAD)   // 1024 float4s per block

__global__ __launch_bounds__(BLOCK_THREADS)
void temp5500_scale_main(const v4f* __restrict__ in,
                         v4f* __restrict__ out) {
    const float sG = 236.0f / 255.0f;
    const float sB = 224.0f / 255.0f;

    // Whole block is inside one channel:
    //   channel = (blockIdx.x * 1024 >> 16) % 3 = (blockIdx.x >> 6) % 3
    unsigned c = (blockIdx.x >> 6) % 3u;
    float s = (c == 0u) ? 1.0f : ((c == 1u) ? sG : sB);

    unsigned base = blockIdx.x * TILE_F4 + threadIdx.x;

    // 4 independent NT b128 loads in flight.
    v4f a = __builtin_nontemporal_load(&in[base + 0 * BLOCK_THREADS]);
    v4f b = __builtin_nontemporal_load(&in[base + 1 * BLOCK_THREADS]);
    v4f d = __builtin_nontemporal_load(&in[base + 2 * BLOCK_THREADS]);
    v4f e = __builtin_nontemporal_load(&in[base + 3 * BLOCK_THREADS]);

    a *= s; b *= s; d *= s; e *= s;

    __builtin_nontemporal_store(a, &out[base + 0 * BLOCK_THREADS]);
    __builtin_nontemporal_store(b, &out[base + 1 * BLOCK_THREADS]);
    __builtin_nontemporal_store(d, &out[base + 2 * BLOCK_THREADS]);
    __builtin_nontemporal_store(e, &out[base + 3 * BLOCK_THREADS]);
}

// Generic scalar tail: elements [start, n), channel = (i / 262144) % 3.
__global__ __launch_bounds__(BLOCK_THREADS)
void temp5500_scale_tail(const float* __restrict__ in,
                         float* __restrict__ out,
                         unsigned start, unsigned n) {
    const float sG = 236.0f / 255.0f;
    const float sB = 224.0f / 255.0f;
    unsigned i = start + blockIdx.x * BLOCK_THREADS + threadIdx.x;
    if (i < n) {
        unsigned c = (i / 262144u) % 3u;
        float s = (c == 0u) ? 1.0f : ((c == 1u) ? sG : sB);
        out[i] = in[i] * s;
    }
}

extern "C" void kernel_launch(void* const* d_in, const int* in_sizes, int n_in,
                              void* d_out, int out_size, void* d_ws, size_t ws_size,
                              hipStream_t stream) {
    (void)n_in; (void)d_ws; (void)ws_size; (void)out_size;

    const float* imgs = (const float*)d_in[0];
    float* out = (float*)d_out;

    unsigned n = (unsigned)in_sizes[0];       // 50,331,648 floats
    unsigned n4 = n / 4u;                     // 12,582,912 float4s (exact)
    unsigned full_blocks = n4 / TILE_F4;      // 12,288 full tiles

    if (full_blocks > 0) {
        temp5500_scale_main<<<full_blocks, BLOCK_THREADS, 0, stream>>>(
            (const v4f*)imgs, (v4f*)out);
    }

    unsigned done = full_blocks * TILE_F4 * 4u;   // elements covered by main
    if (done < n) {
        unsigned rem = n - done;
        unsigned tail_blocks = (rem + BLOCK_THREADS - 1u) / BLOCK_THREADS;
        temp5500_scale_tail<<<tail_blocks, BLOCK_THREADS, 0, stream>>>(
            imgs, out, done, n);
    }
}